// KascadeAnchorAttention_28312424415932
// MI455X (gfx1250) — compile-verified
//
#include <hip/hip_runtime.h>
#include <hip/hip_bf16.h>

// ---------------------------------------------------------------------------
// KascadeAnchorAttention for MI455X (gfx1250, wave32, WMMA + TDM).
// fp32 end-to-end with V_WMMA_F32_16X16X4_F32 (matches fp32 reference).
// All 2D tile staging (X/W chunks, K/V tiles) done by the Tensor Data Mover
// (tensor_load_to_lds) with LDS padding for conflict-free layouts, double
// buffered on TENSORcnt.
//
// B=2 S=2048 H=16 D=64 DM=1024 TILE=16 NT=128 TOPK=8
// Workspace: Q[B,H,S,D] | K[B,H,S,D] | V[B,S,H,D] | AO[B,S,DM] = 64 MB fp32
// ---------------------------------------------------------------------------

#define BK   2
#define SK   2048
#define HK   16
#define DK   64
#define DMK  1024
#define NTK  128
#define TOPKK 8

typedef __attribute__((ext_vector_type(2))) float v2f;
typedef __attribute__((ext_vector_type(8))) float v8f;
typedef __attribute__((ext_vector_type(4))) unsigned int u32x4;
typedef __attribute__((ext_vector_type(4))) int i32x4;
typedef __attribute__((ext_vector_type(8))) int i32x8;

static __device__ __forceinline__ v8f wmma4(v2f a, v2f b, v8f c) {
  // D(16x16,f32) = A(16x4,f32) x B(4x16,f32) + C
  return __builtin_amdgcn_wmma_f32_16x16x4_f32(false, a, false, b,
                                               (short)0, c, false, false);
}

// ---------------------------------------------------------------------------
// TDM 2D tile load: rows x dim0 fp32 tile, global row stride (elements),
// into LDS at lds_off with padding (pad codes per CDNA5 D# encoding:
// interval code v -> pad after 2^(v+1) DWORDs; amount code a -> a+1 DWORDs).
// ---------------------------------------------------------------------------
static __device__ __forceinline__ void tdm_load_2d(
    unsigned int lds_off, const float* gptr,
    unsigned int dim0, unsigned int rows,
    unsigned long long row_stride_elems,
    unsigned int pad_interval_code, unsigned int pad_amount_code) {
  unsigned long long ga = (unsigned long long)(uintptr_t)gptr;

  u32x4 g0;
  g0[0] = 1u;                                       // count=1 (valid), user mode
  g0[1] = lds_off;                                  // lds_addr (bytes)
  g0[2] = (unsigned int)ga;                         // global_addr[31:0]
  g0[3] = (unsigned int)(ga >> 32) | (2u << 30);    // global_addr[56:32] | type=2

  union { unsigned long long q[4]; int w[8]; } g1u;
  // q0: mask=0 | data_size(4B)=2<<16 | pad_enable<<20 | pint<<22 | pamt<<25
  //     | atomic_barrier_addr=0 | tensor_dim0[15:0]<<48
  g1u.q[0] = (2ull << 16) | (1ull << 20) |
             ((unsigned long long)pad_interval_code << 22) |
             ((unsigned long long)pad_amount_code << 25) |
             ((unsigned long long)(dim0 & 0xFFFFu) << 48);
  // q1: tensor_dim0[31:16] | tensor_dim1<<16 | tile_dim0<<48
  g1u.q[1] = ((unsigned long long)dim0 >> 16) |
             ((unsigned long long)rows << 16) |
             ((unsigned long long)(dim0 & 0xFFFFu) << 48);
  // q2: tile_dim1 | tile_dim2(=0)<<16 | tensor_dim0_stride[31:0]<<32
  g1u.q[2] = (unsigned long long)(rows & 0xFFFFu) |
             ((row_stride_elems & 0xFFFFFFFFull) << 32);
  // q3: tensor_dim0_stride[47:32] | tensor_dim1_stride(=0)<<16
  g1u.q[3] = (row_stride_elems >> 32) & 0xFFFFull;

  i32x8 g1;
  #pragma unroll
  for (int i = 0; i < 8; ++i) g1[i] = g1u.w[i];
  i32x4 z4 = {0, 0, 0, 0};
#if __has_include(<hip/amd_detail/amd_gfx1250_TDM.h>)
  i32x8 z8 = {0, 0, 0, 0, 0, 0, 0, 0};
  __builtin_amdgcn_tensor_load_to_lds(g0, g1, z4, z4, z8, 0);
#else
  __builtin_amdgcn_tensor_load_to_lds(g0, g1, z4, z4, 0);
#endif
}

template <int N>
static __device__ __forceinline__ void tdm_wait() {
  __builtin_amdgcn_s_wait_tensorcnt(N);
  asm volatile("" ::: "memory");   // keep ds reads behind the wait
}

static __device__ __forceinline__ unsigned int lds_off_of(const void* p) {
  return (unsigned int)(uintptr_t)p;
}

// ---------------------------------------------------------------------------
// Kernel 1: QKV projection.  One wave computes one 16x16 output tile.
// grid.x = (B*S)/16 = 256 row tiles, grid.y = 192 (64 col tiles x {Q,K,V}).
// X chunks (16x64, stride DM) and W chunks (64x16, stride DM) streamed by
// TDM, double buffered on TENSORcnt.
// ---------------------------------------------------------------------------
__global__ __launch_bounds__(32) void qkv_proj_kernel(
    const float* __restrict__ x,
    const float* __restrict__ wq, const float* __restrict__ wk,
    const float* __restrict__ wv,
    float* __restrict__ Q, float* __restrict__ K, float* __restrict__ V) {
  __shared__ float Xs[2 * 16 * 68];
  __shared__ float Ws[2 * 64 * 17];

  const int lane = threadIdx.x;
  const int half = lane >> 4;
  const int l16  = lane & 15;

  const int m0    = blockIdx.x * 16;
  const int ny    = blockIdx.y;
  const int which = ny >> 6;                  // 0=Q 1=K 2=V
  const int n0    = (ny & 63) * 16;
  const float* __restrict__ w = (which == 0) ? wq : ((which == 1) ? wk : wv);

  v8f acc = {0.f, 0.f, 0.f, 0.f, 0.f, 0.f, 0.f, 0.f};

  // Prime chunk 0.
  tdm_load_2d(lds_off_of(&Xs[0]), x + (size_t)m0 * DMK, 64, 16, DMK, 5, 3);
  tdm_load_2d(lds_off_of(&Ws[0]), w + n0, 16, 64, DMK, 3, 0);

  for (int c = 0; c < 16; ++c) {
    const int k0 = c * 64;
    if (c < 15) {   // stream next chunk into the other buffer
      int nb = (c + 1) & 1;
      tdm_load_2d(lds_off_of(&Xs[nb * 16 * 68]),
                  x + (size_t)m0 * DMK + k0 + 64, 64, 16, DMK, 5, 3);
      tdm_load_2d(lds_off_of(&Ws[nb * 64 * 17]),
                  w + (size_t)(k0 + 64) * DMK + n0, 16, 64, DMK, 3, 0);
      __builtin_prefetch(x + (size_t)m0 * DMK + k0 + 128, 0, 1);
      tdm_wait<2>();   // in-order TDM: older pair complete
    } else {
      tdm_wait<0>();
    }

    const float* Xb = &Xs[(c & 1) * 16 * 68];
    const float* Wb = &Ws[(c & 1) * 64 * 17];
    // Stage all fragments, then run WMMAs back-to-back.
    v2f af[16], bf[16];
    #pragma unroll
    for (int t = 0; t < 16; ++t) {
      int kk = 4 * t;
      af[t].x = Xb[l16 * 68 + kk + 2 * half];
      af[t].y = Xb[l16 * 68 + kk + 1 + 2 * half];
      bf[t].x = Wb[(kk + 2 * half) * 17 + l16];
      bf[t].y = Wb[(kk + 1 + 2 * half) * 17 + l16];
    }
    #pragma unroll
    for (int t = 0; t < 16; ++t) acc = wmma4(af[t], bf[t], acc);
  }

  // Write out.  D layout: VGPR i -> row M = i + 8*half, col N = l16.
  #pragma unroll
  for (int i = 0; i < 8; ++i) {
    int M   = i + 8 * half;
    int r   = m0 + M;                 // flattened b*S + s
    int b   = r >> 11;
    int s   = r & (SK - 1);
    int col = n0 + l16;
    float val = acc[i];
    if (which == 2) {
      V[(size_t)r * DMK + col] = val;               // V[b,s,h,d] flat
    } else {
      int h = col >> 6, d = col & 63;
      float* dst = (which == 0) ? Q : K;
      dst[(((size_t)(b * HK + h)) * SK + s) * DK + d] = val;
    }
  }
}

// ---------------------------------------------------------------------------
// Kernel 2: RoPE on Q and K in [B,H,S,D].  Pairs (f, f+32), f in [0,32).
// ---------------------------------------------------------------------------
__global__ __launch_bounds__(256) void rope_kernel(
    float* __restrict__ Q, float* __restrict__ K,
    const float* __restrict__ fc) {
  size_t t = (size_t)blockIdx.x * 256 + threadIdx.x;
  float* buf = (blockIdx.y == 0) ? Q : K;
  int f      = (int)(t & 31);
  size_t row = t >> 5;
  int s      = (int)(row & (SK - 1));
  float c  = fc[(size_t)s * 64 + 2 * f];
  float si = fc[(size_t)s * 64 + 2 * f + 1];
  float x0 = buf[row * DK + f];
  float x1 = buf[row * DK + f + 32];
  buf[row * DK + f]      = x0 * c - x1 * si;
  buf[row * DK + f + 32] = x1 * c + x0 * si;
}

// ---------------------------------------------------------------------------
// Kernel 3: flash attention + tile-max top-k.  One wave per (b,h,q-tile).
// K/V tiles streamed by TDM, double buffered.  grid.x = B*H*(S/16) = 4096.
// ---------------------------------------------------------------------------
__global__ __launch_bounds__(32) void attn_kernel(
    const float* __restrict__ Q, const float* __restrict__ K,
    const float* __restrict__ V, float* __restrict__ AO,
    int* __restrict__ IDX) {
  __shared__ float Qs[16 * 68];
  __shared__ float Ks[2 * 16 * 68];
  __shared__ float Vs[2 * 16 * 68];
  __shared__ float Ps[16 * 20];
  __shared__ float Ts[16 * NTK];

  const int lane = threadIdx.x;
  const int half = lane >> 4;
  const int l16  = lane & 15;

  const int bx = blockIdx.x;
  const int qt = bx & (NTK - 1);
  const int h  = (bx >> 7) & (HK - 1);
  const int b  = bx >> 11;
  const int q0 = qt * 16;

  const float* __restrict__ Qb = Q + ((size_t)(b * HK + h) * SK) * DK;
  const float* __restrict__ Kb = K + ((size_t)(b * HK + h) * SK) * DK;
  const float* __restrict__ Vb = V + (size_t)b * SK * DMK + h * DK;

  // Stage Q tile (TDM) and prime K/V tile 0.
  tdm_load_2d(lds_off_of(&Qs[0]), Qb + (size_t)q0 * DK, 64, 16, DK, 5, 3);
  tdm_load_2d(lds_off_of(&Ks[0]), Kb, 64, 16, DK, 5, 3);
  tdm_load_2d(lds_off_of(&Vs[0]), Vb, 64, 16, DMK, 5, 3);

  // Tiles beyond qt are fully masked -> tile-max logit == -1e10 (ref tie).
  #pragma unroll
  for (int j = 0; j < 64; ++j) Ts[lane + 32 * j] = -1e10f;

  float m[8], lsum[8];
  #pragma unroll
  for (int i = 0; i < 8; ++i) { m[i] = -3.0e38f; lsum[i] = 0.0f; }
  v8f o0 = {0.f,0.f,0.f,0.f,0.f,0.f,0.f,0.f};
  v8f o1 = o0, o2 = o0, o3 = o0;

  // Wait for Q + K/V pair 0, then hoist Q fragments (loop invariant).
  tdm_wait<0>();
  v2f qa[16];
  #pragma unroll
  for (int t = 0; t < 16; ++t) {
    int kk = 4 * t;
    qa[t].x = Qs[l16 * 68 + kk + 2 * half];
    qa[t].y = Qs[l16 * 68 + kk + 1 + 2 * half];
  }

  for (int kt = 0; kt <= qt; ++kt) {
    const int kr = kt * 16;
    if (kt < qt) {  // stream next K/V tile into the other buffer
      int nb = (kt + 1) & 1;
      tdm_load_2d(lds_off_of(&Ks[nb * 16 * 68]),
                  Kb + (size_t)(kr + 16) * DK, 64, 16, DK, 5, 3);
      tdm_load_2d(lds_off_of(&Vs[nb * 16 * 68]),
                  Vb + (size_t)(kr + 16) * DMK, 64, 16, DMK, 5, 3);
      tdm_wait<2>();   // in-order TDM: pair(kt) complete
    } else {
      tdm_wait<0>();
    }

    const float* Kl = &Ks[(kt & 1) * 16 * 68];
    const float* Vl = &Vs[(kt & 1) * 16 * 68];

    // S_tile = Q (16x64) @ K^T (64x16): stage B frags, 16 WMMAs back-to-back.
    v2f kbf[16];
    #pragma unroll
    for (int t = 0; t < 16; ++t) {
      int kk = 4 * t;
      kbf[t].x = Kl[l16 * 68 + kk + 2 * half];      // B[k][N] = K[N][k]
      kbf[t].y = Kl[l16 * 68 + kk + 1 + 2 * half];
    }
    v8f sf = {0.f,0.f,0.f,0.f,0.f,0.f,0.f,0.f};
    #pragma unroll
    for (int t = 0; t < 16; ++t) sf = wmma4(qa[t], kbf[t], sf);

    // Scale by 1/sqrt(D)=0.125, apply causal mask with -1e10 (as reference).
    #pragma unroll
    for (int i = 0; i < 8; ++i) {
      int gq = q0 + i + 8 * half;
      int gk = kr + l16;
      float v = sf[i] * 0.125f;
      sf[i] = (gk > gq) ? -1e10f : v;
    }

    // Row max across the 16 lanes of each half.
    float rm[8];
    #pragma unroll
    for (int i = 0; i < 8; ++i) {
      float v = sf[i];
      #pragma unroll
      for (int d = 1; d < 16; d <<= 1) v = fmaxf(v, __shfl_xor(v, d, 32));
      rm[i] = v;
    }
    if (l16 == 0) {
      #pragma unroll
      for (int i = 0; i < 8; ++i) Ts[(i + 8 * half) * NTK + kt] = rm[i];
    }

    // Online softmax update.
    float alpha[8];
    #pragma unroll
    for (int i = 0; i < 8; ++i) {
      float mn = fmaxf(m[i], rm[i]);
      alpha[i] = __expf(m[i] - mn);
      m[i] = mn;
      sf[i] = __expf(sf[i] - mn);
    }
    #pragma unroll
    for (int i = 0; i < 8; ++i) {
      float v = sf[i];
      #pragma unroll
      for (int d = 1; d < 16; d <<= 1) v += __shfl_xor(v, d, 32);
      lsum[i] = lsum[i] * alpha[i] + v;
    }

    // Route P through LDS into WMMA A-layout; rescale O; O += P @ V.
    #pragma unroll
    for (int i = 0; i < 8; ++i) Ps[(i + 8 * half) * 20 + l16] = sf[i];
    #pragma unroll
    for (int i = 0; i < 8; ++i) {
      o0[i] *= alpha[i]; o1[i] *= alpha[i];
      o2[i] *= alpha[i]; o3[i] *= alpha[i];
    }
    v2f pa[4], vbf[4][4];
    #pragma unroll
    for (int t = 0; t < 4; ++t) {
      int kk = 4 * t;
      pa[t].x = Ps[l16 * 20 + kk + 2 * half];
      pa[t].y = Ps[l16 * 20 + kk + 1 + 2 * half];
      int r0 = (kk + 2 * half) * 68, r1 = (kk + 1 + 2 * half) * 68;
      #pragma unroll
      for (int n = 0; n < 4; ++n) {
        vbf[t][n].x = Vl[r0 + 16 * n + l16];
        vbf[t][n].y = Vl[r1 + 16 * n + l16];
      }
    }
    #pragma unroll
    for (int t = 0; t < 4; ++t) {
      o0 = wmma4(pa[t], vbf[t][0], o0);
      o1 = wmma4(pa[t], vbf[t][1], o1);
      o2 = wmma4(pa[t], vbf[t][2], o2);
      o3 = wmma4(pa[t], vbf[t][3], o3);
    }
  }

  // Epilogue: normalize rows and write AO[b,s,h,d].
  float inv[8];
  #pragma unroll
  for (int i = 0; i < 8; ++i) inv[i] = 1.0f / lsum[i];
  float* __restrict__ aob = AO + (((size_t)(b * SK + q0)) * HK + h) * DK;
  #pragma unroll
  for (int i = 0; i < 8; ++i) {
    size_t ro = (size_t)(i + 8 * half) * DMK;
    aob[ro + 0  + l16] = o0[i] * inv[i];
    aob[ro + 16 + l16] = o1[i] * inv[i];
    aob[ro + 32 + l16] = o2[i] * inv[i];
    aob[ro + 48 + l16] = o3[i] * inv[i];
  }

  // Top-8 tile indices per query row (stable first-occurrence tie-break,
  // matching jax.lax.top_k on the monotone-equivalent tile-max logits).
  __syncthreads();
  if (lane < 16) {
    float* trow = Ts + lane * NTK;
    int* __restrict__ ob = IDX + ((size_t)(b * HK + h) * SK + q0 + lane) * TOPKK;
    for (int t = 0; t < TOPKK; ++t) {
      float best = -3.4e38f;
      int bi = 0;
      for (int j = 0; j < NTK; ++j) {
        float v = trow[j];
        if (v > best) { best = v; bi = j; }
      }
      ob[t] = bi;
      trow[bi] = -3.4e38f;
    }
  }
}

// ---------------------------------------------------------------------------
// Kernel 4: output projection AO[4096,1024] @ wo[1024,1024] -> out.
// ---------------------------------------------------------------------------
__global__ __launch_bounds__(32) void out_proj_kernel(
    const float* __restrict__ AO, const float* __restrict__ wo,
    float* __restrict__ out) {
  __shared__ float Xs[2 * 16 * 68];
  __shared__ float Ws[2 * 64 * 17];

  const int lane = threadIdx.x;
  const int half = lane >> 4;
  const int l16  = lane & 15;
  const int m0 = blockIdx.x * 16;
  const int n0 = blockIdx.y * 16;

  v8f acc = {0.f, 0.f, 0.f, 0.f, 0.f, 0.f, 0.f, 0.f};

  tdm_load_2d(lds_off_of(&Xs[0]), AO + (size_t)m0 * DMK, 64, 16, DMK, 5, 3);
  tdm_load_2d(lds_off_of(&Ws[0]), wo + n0, 16, 64, DMK, 3, 0);

  for (int c = 0; c < 16; ++c) {
    const int k0 = c * 64;
    if (c < 15) {
      int nb = (c + 1) & 1;
      tdm_load_2d(lds_off_of(&Xs[nb * 16 * 68]),
                  AO + (size_t)m0 * DMK + k0 + 64, 64, 16, DMK, 5, 3);
      tdm_load_2d(lds_off_of(&Ws[nb * 64 * 17]),
                  wo + (size_t)(k0 + 64) * DMK + n0, 16, 64, DMK, 3, 0);
      tdm_wait<2>();
    } else {
      tdm_wait<0>();
    }

    const float* Xb = &Xs[(c & 1) * 16 * 68];
    const float* Wb = &Ws[(c & 1) * 64 * 17];
    v2f af[16], bf[16];
    #pragma unroll
    for (int t = 0; t < 16; ++t) {
      int kk = 4 * t;
      af[t].x = Xb[l16 * 68 + kk + 2 * half];
      af[t].y = Xb[l16 * 68 + kk + 1 + 2 * half];
      bf[t].x = Wb[(kk + 2 * half) * 17 + l16];
      bf[t].y = Wb[(kk + 1 + 2 * half) * 17 + l16];
    }
    #pragma unroll
    for (int t = 0; t < 16; ++t) acc = wmma4(af[t], bf[t], acc);
  }

  #pragma unroll
  for (int i = 0; i < 8; ++i) {
    out[(size_t)(m0 + i + 8 * half) * DMK + n0 + l16] = acc[i];
  }
}

// ---------------------------------------------------------------------------
extern "C" void kernel_launch(void* const* d_in, const int* in_sizes, int n_in,
                              void* d_out, int out_size, void* d_ws,
                              size_t ws_size, hipStream_t stream) {
  (void)in_sizes; (void)n_in; (void)out_size; (void)ws_size;
  const float* x  = (const float*)d_in[0];
  const float* wq = (const float*)d_in[1];
  const float* wk = (const float*)d_in[2];
  const float* wv = (const float*)d_in[3];
  const float* wo = (const float*)d_in[4];
  const float* fc = (const float*)d_in[5];   // complex64 interleaved

  float* out = (float*)d_out;
  int* idx_out = (int*)(out + (size_t)BK * SK * DMK);

  const size_t qkv_elems = (size_t)BK * HK * SK * DK;   // 4 Mi floats
  float* Q  = (float*)d_ws;
  float* K  = Q + qkv_elems;
  float* V  = K + qkv_elems;
  float* AO = V + qkv_elems;

  qkv_proj_kernel<<<dim3((BK * SK) / 16, 192), 32, 0, stream>>>(
      x, wq, wk, wv, Q, K, V);
  rope_kernel<<<dim3((BK * HK * SK * 32) / 256, 2), 256, 0, stream>>>(Q, K, fc);
  attn_kernel<<<dim3(BK * HK * (SK / 16)), 32, 0, stream>>>(Q, K, V, AO,
                                                            idx_out);
  out_proj_kernel<<<dim3((BK * SK) / 16, DMK / 16), 32, 0, stream>>>(AO, wo,
                                                                     out);
}